// Router_44272522887247
// MI455X (gfx1250) — compile-verified
//
#include <hip/hip_runtime.h>
#include <math.h>

// ---------------- problem constants (static per reference) ----------------
#define NTOK    8192          // B*T = 2*4096
#define DDIM    1024
#define NEXP    16
#define CAP     512           // ceil(1.0 * 8192 / 16)
#define TOK_BLK 256
#define NBLK    (NTOK / TOK_BLK)   // 32

typedef __attribute__((ext_vector_type(2))) float v2f;
typedef __attribute__((ext_vector_type(4))) float v4f;
typedef __attribute__((ext_vector_type(8))) float v8f;

// ---------------------------------------------------------------------------
// Kernel 1: zero-fill the two sparse [N,E,cap] outputs (537 MB -> dominates
// runtime).  16B non-temporal streaming stores; 537MB >> 192MB L2 so NT is
// the right temporal hint.
// ---------------------------------------------------------------------------
__global__ void __launch_bounds__(256) k_fill0(v4f* __restrict__ p, size_t n4) {
    size_t i      = (size_t)blockIdx.x * blockDim.x + threadIdx.x;
    size_t stride = (size_t)gridDim.x * blockDim.x;
    v4f z = {};
    for (; i < n4; i += stride)
        __builtin_nontemporal_store(z, p + i);
}

// ---------------------------------------------------------------------------
// Kernel 2: logits[n,e] = sum_k x[n,k]*gw[e,k] via V_WMMA_F32_16X16X4_F32.
// One wave per 16-token tile (E=16 == WMMA N exactly); 256 chained WMMAs over
// K=1024.  gate_w (64 KB) staged in LDS once per 256-thread block (8 waves).
//
// Fragment layouts (ISA 7.12.2):
//   A 16x4 f32, 2 VGPRs: lane L -> M = L&15, K = (L>>4)*2 + j
//   B 4x16 f32, 2 VGPRs: lane L -> N = L&15, K = (L>>4)*2 + j  (B[k][n]=gw[n][k])
//   C/D 16x16 f32, 8 VGPRs: lane L -> N = L&15, M = r + (L>>4)*8
// ---------------------------------------------------------------------------
__global__ void __launch_bounds__(256) k_wmma_logits(const float* __restrict__ x,
                                                     const float* __restrict__ gw,
                                                     float* __restrict__ logits) {
    __shared__ float lgw[NEXP * DDIM];            // 64 KB of the 320 KB WGP LDS
    for (int i = threadIdx.x; i < NEXP * DDIM; i += 256) lgw[i] = gw[i];
    __syncthreads();

    const int lane  = threadIdx.x & 31;
    const int wave  = threadIdx.x >> 5;           // wave32 (gfx1250)
    const int tile  = blockIdx.x * 8 + wave;      // 16-token tile id
    const int half  = lane >> 4;                  // K sub-phase (0 or 1)
    const int idx16 = lane & 15;                  // M for A, N (expert) for B

    const float* ap = x   + (size_t)(tile * 16 + idx16) * DDIM + half * 2;
    const float* bp = lgw + (size_t)idx16 * DDIM + half * 2;

    v8f acc = {};
#pragma unroll 8
    for (int k = 0; k < DDIM; k += 4) {
        v2f a = *(const v2f*)(ap + k);            // global_load_b64, coalesced
        v2f b = *(const v2f*)(bp + k);            // ds_load_b64
        acc = __builtin_amdgcn_wmma_f32_16x16x4_f32(
            /*neg_a=*/false, a, /*neg_b=*/false, b,
            /*c_mod=*/(short)0, acc, /*reuse_a=*/false, /*reuse_b=*/false);
    }

    float* op = logits + (size_t)tile * 16 * NEXP;
#pragma unroll
    for (int r = 0; r < 8; ++r)
        op[(half * 8 + r) * NEXP + idx16] = acc[r];
}

// ---------------------------------------------------------------------------
// Kernel 3: per-token softmax / argmax / logsumexp, per-block expert histogram
// and deterministic (fixed-order) per-block partial sums for the aux loss.
// probs written straight to d_out.  No float atomics -> bit-stable replays.
// ---------------------------------------------------------------------------
__global__ void __launch_bounds__(TOK_BLK) k_softmax(const float* __restrict__ logits,
        float* __restrict__ probs, int* __restrict__ eidx, float* __restrict__ wts,
        int* __restrict__ counts, float* __restrict__ blockP, float* __restrict__ blockZ) {
    __shared__ float sp[TOK_BLK * NEXP];          // 16 KB
    __shared__ float slse2[TOK_BLK];
    __shared__ int   hist[NEXP];

    const int tid = threadIdx.x;
    const int n   = blockIdx.x * TOK_BLK + tid;
    if (tid < NEXP) hist[tid] = 0;
    __syncthreads();

    float l[NEXP];
    const float* lp = logits + (size_t)n * NEXP;
#pragma unroll
    for (int e = 0; e < NEXP; ++e) l[e] = lp[e];

    float mx = l[0]; int amax = 0;                // first-index tie-break == jnp.argmax
#pragma unroll
    for (int e = 1; e < NEXP; ++e) if (l[e] > mx) { mx = l[e]; amax = e; }

    float s = 0.f;
#pragma unroll
    for (int e = 0; e < NEXP; ++e) { l[e] = __expf(l[e] - mx); s += l[e]; }
    const float inv = 1.f / s;

    float* pp = probs + (size_t)n * NEXP;
#pragma unroll
    for (int e = 0; e < NEXP; ++e) { float p = l[e] * inv; pp[e] = p; sp[tid * NEXP + e] = p; }

    eidx[n] = amax;
    wts[n]  = l[amax] * inv;
    atomicAdd(&hist[amax], 1);                    // integer: deterministic
    const float lse = mx + __logf(s);
    slse2[tid] = lse * lse;
    __syncthreads();

    if (tid < NEXP) {
        counts[blockIdx.x * NEXP + tid] = hist[tid];
        float acc = 0.f;
        for (int t = 0; t < TOK_BLK; ++t) acc += sp[t * NEXP + tid];   // fixed order
        blockP[blockIdx.x * NEXP + tid] = acc;
    }
    if (tid == 0) {
        float z = 0.f;
        for (int t = 0; t < TOK_BLK; ++t) z += slse2[t];               // fixed order
        blockZ[blockIdx.x] = z;
    }
}

// ---------------------------------------------------------------------------
// Kernel 4: exclusive scan of per-block expert counts (token-order preserving)
// + aux_loss finalization.  1 block, trivially cheap (E=16, NBLK=32).
// ---------------------------------------------------------------------------
__global__ void k_scan_aux(const int* __restrict__ counts, int* __restrict__ offsets,
        const float* __restrict__ blockP, const float* __restrict__ blockZ,
        float* __restrict__ aux_out) {
    __shared__ float fp[NEXP];
    const int e = threadIdx.x;
    if (e < NEXP) {
        int run = 0; float psum = 0.f;
        for (int b = 0; b < NBLK; ++b) {
            offsets[b * NEXP + e] = run;
            run  += counts[b * NEXP + e];
            psum += blockP[b * NEXP + e];
        }
        const float fi = (float)run / (float)NTOK;
        const float pi = psum / (float)NTOK;
        fp[e] = fi * pi;
    }
    __syncthreads();
    if (e == 0) {
        float s = 0.f;
        for (int i = 0; i < NEXP; ++i) s += fp[i];
        float z = 0.f;
        for (int b = 0; b < NBLK; ++b) z += blockZ[b];
        aux_out[0] = 0.01f * (float)NEXP * s + 0.001f * (z / (float)NTOK);
    }
}

// ---------------------------------------------------------------------------
// Kernel 5: slot = block-base offset + in-block rank (token order), then
// scatter the <=8192 nonzeros into the zero-filled outputs (capacity drop
// for slot >= 512).
// ---------------------------------------------------------------------------
__global__ void __launch_bounds__(TOK_BLK) k_scatter(const int* __restrict__ eidx,
        const float* __restrict__ wts, const int* __restrict__ offsets,
        float* __restrict__ dmask, float* __restrict__ cmb) {
    __shared__ int se[TOK_BLK];
    const int tid = threadIdx.x;
    const int n   = blockIdx.x * TOK_BLK + tid;
    const int e   = eidx[n];
    se[tid] = e;
    __syncthreads();
    int rank = 0;
    for (int j = 0; j < tid; ++j) rank += (se[j] == e) ? 1 : 0;
    const int slot = offsets[blockIdx.x * NEXP + e] + rank;
    if (slot < CAP) {
        const size_t idx = (((size_t)n * NEXP) + e) * CAP + slot;
        dmask[idx] = 1.0f;
        cmb[idx]   = wts[n];
    }
}

// ---------------------------------------------------------------------------
extern "C" void kernel_launch(void* const* d_in, const int* in_sizes, int n_in,
                              void* d_out, int out_size, void* d_ws, size_t ws_size,
                              hipStream_t stream) {
    const float* x  = (const float*)d_in[0];   // [2,4096,1024] f32
    const float* gw = (const float*)d_in[1];   // [16,1024] f32
    float* out = (float*)d_out;
    char*  ws  = (char*)d_ws;

    // workspace layout (~600 KB)
    float* logits  = (float*)(ws);                              // N*E f32   (512 KB)
    int*   eidx    = (int*)  (ws + 524288);                     // N  i32    (32 KB)
    float* wts     = (float*)(ws + 524288 + 32768);             // N  f32    (32 KB)
    int*   counts  = (int*)  (ws + 524288 + 65536);             // NBLK*E    (2 KB)
    int*   offsets = (int*)  (ws + 524288 + 65536 + 2048);      // NBLK*E    (2 KB)
    float* blockP  = (float*)(ws + 524288 + 65536 + 4096);      // NBLK*E    (2 KB)
    float* blockZ  = (float*)(ws + 524288 + 65536 + 4096 + 2048); // NBLK    (128 B)

    const size_t NEC = (size_t)NTOK * NEXP * CAP;               // 67,108,864
    float* dmask = out;                 // [N,E,cap]
    float* cmb   = out + NEC;           // [N,E,cap]
    float* auxp  = out + 2 * NEC;       // scalar
    float* probs = out + 2 * NEC + 1;   // [N,E]

    // 1) streaming zero-fill of both sparse outputs (bandwidth floor ~23 us)
    k_fill0<<<4096, 256, 0, stream>>>((v4f*)out, (2 * NEC) / 4);
    // 2) logits GEMM on the WMMA path (512 tiles, 8 waves/block -> 64 blocks)
    k_wmma_logits<<<NTOK / (16 * 8), 256, 0, stream>>>(x, gw, logits);
    // 3) softmax / argmax / histograms / deterministic partials
    k_softmax<<<NBLK, TOK_BLK, 0, stream>>>(logits, probs, eidx, wts, counts, blockP, blockZ);
    // 4) cross-block exclusive scan + aux loss
    k_scan_aux<<<1, 32, 0, stream>>>(counts, offsets, blockP, blockZ, auxp);
    // 5) sparse scatter of dispatch/combine nonzeros
    k_scatter<<<NBLK, TOK_BLK, 0, stream>>>(eidx, wts, offsets, dmask, cmb);
}